// LocalSelfAttention_29111288332435
// MI455X (gfx1250) — compile-verified
//
#include <hip/hip_runtime.h>
#include <hip/hip_bf16.h>

typedef float v2f __attribute__((ext_vector_type(2)));
typedef float v8f __attribute__((ext_vector_type(8)));

// V_WMMA_F32_16X16X4_F32 : A 16x4 f32 (2 VGPR), B 4x16 f32 (2 VGPR), C/D 16x16 f32 (8 VGPR)
__device__ __forceinline__ v8f wmma_f32(v2f a, v2f b, v8f c) {
    return __builtin_amdgcn_wmma_f32_16x16x4_f32(false, a, false, b, (short)0, c, false, false);
}

// ---------------------------------------------------------------------------
// Kernel 1/4: C[M x Nc] = A[M x K] @ Bw[K x Nc] (+ bias). One wave = 16x64 tile.
// A-frag: lane L (half=L/16,l16=L&15): row=l16, K = k0 + 2*half + {0,1}  (float2)
// B-frag: lane L: col = n0+l16,        K = k0 + 2*half + {0,1}
// D     : vgpr r: row = r + 8*half, col = n0 + l16
// ---------------------------------------------------------------------------
__global__ __launch_bounds__(256)
void gemm16x64_f32wmma(const float* __restrict__ A, const float* __restrict__ Bw,
                       const float* __restrict__ bias, float* __restrict__ C,
                       int K, int Nc)
{
    const int lane = threadIdx.x & 31;
    const int wave = threadIdx.x >> 5;
    const int half = lane >> 4;
    const int l16  = lane & 15;
    const int row0 = (blockIdx.y * 8 + wave) * 16;
    const int n0   = blockIdx.x * 64;

    v8f c[4] = {};
    const float* aptr = A + (size_t)(row0 + l16) * K + half * 2;
    for (int k0 = 0; k0 < K; k0 += 4) {
        v2f a = *(const v2f*)(aptr + k0);
        const int kb = k0 + half * 2;
#pragma unroll
        for (int t = 0; t < 4; ++t) {
            const int col = n0 + t * 16 + l16;
            v2f b;
            b.x = Bw[(size_t)kb * Nc + col];
            b.y = Bw[(size_t)(kb + 1) * Nc + col];
            c[t] = wmma_f32(a, b, c[t]);
        }
    }
#pragma unroll
    for (int t = 0; t < 4; ++t) {
        const int col = n0 + t * 16 + l16;
        const float bv = bias ? bias[col] : 0.0f;
#pragma unroll
        for (int r = 0; r < 8; ++r) {
            const int m = r + half * 8;
            C[(size_t)(row0 + m) * Nc + col] = c[t][r] + bv;
        }
    }
}

// ---------------------------------------------------------------------------
// Kernel 2: RoPE on q,k + scatter qkv (B*N, 3, h, d) -> Q/K/V (B, h, N, d).
// Q is pre-scaled by 1/sqrt(d). One thread per (b, n, h, pair p in 0..15).
// ---------------------------------------------------------------------------
__global__ void rope_scatter(const float* __restrict__ qkv,
                             float* __restrict__ Q, float* __restrict__ Kk,
                             float* __restrict__ V, int BN, int N)
{
    int idx = blockIdx.x * blockDim.x + threadIdx.x;
    if (idx >= BN * 8 * 16) return;
    const int p  = idx & 15;
    const int h  = (idx >> 4) & 7;
    const int bn = idx >> 7;
    const int n  = bn % N;
    const int b  = bn / N;

    const float* src = qkv + (size_t)bn * 768 + h * 32 + 2 * p;
    const float q0 = src[0],   q1 = src[1];
    const float k0 = src[256], k1 = src[257];
    const float v0 = src[512], v1 = src[513];

    // inv_freq = 10000^(-2p/32) = exp(-p * ln(10000)/16)
    const float inv = __expf(-(float)p * (9.210340371976184f / 16.0f));
    const float ang = (float)n * inv;
    const float cc = __cosf(ang), ss = __sinf(ang);
    const float scale = 0.17677669529663687f; // 1/sqrt(32)

    const size_t dst = ((size_t)(b * 8 + h) * N + n) * 32 + 2 * p;
    Q[dst]     = (q0 * cc - q1 * ss) * scale;
    Q[dst + 1] = (q1 * cc + q0 * ss) * scale;
    Kk[dst]     = k0 * cc - k1 * ss;
    Kk[dst + 1] = k1 * cc + k0 * ss;
    V[dst]     = v0;
    V[dst + 1] = v1;
}

// ---------------------------------------------------------------------------
// Kernel 3: 7x7 neighborhood attention via WMMA.
// Block = 1 wave = 16 queries (i, j0..j0+15), one (b,h).
// Shared key window: 7 rows x 22 cols = 154 keys, padded to 160 (10 tiles).
// S = Q(16x32) @ K^T(32x160); masked softmax; O = P(16x160) @ V(160x32).
// ---------------------------------------------------------------------------
__global__ __launch_bounds__(32)
void natten_wmma(const float* __restrict__ Q, const float* __restrict__ Kk,
                 const float* __restrict__ V, float* __restrict__ O)
{
    const int Wd = 56, Npix = 3136;
    const int jt = blockIdx.x;        // 0..3  (16-query column groups)
    const int i  = blockIdx.y;        // 0..55
    const int bh = blockIdx.z;        // 0..15 (b*8+h)
    const int lane = threadIdx.x, half = lane >> 4, l16 = lane & 15;
    const int j0 = jt * 16;

    const int rstart = min(max(i - 3, 0), 49);
    const int cstart = min(max(j0 - 3, 0), 49);

    const float* Qb = Q  + (size_t)bh * Npix * 32;
    const float* Kb = Kk + (size_t)bh * Npix * 32;
    const float* Vb = V  + (size_t)bh * Npix * 32;

    // ---- load Q tile as 8 A-fragments (d = 32 -> 8 k-steps of 4) ----
    v2f a[8];
    const int jq_l = min(j0 + l16, Wd - 1);        // clamp phantom queries
    const float* qrow = Qb + (size_t)(i * Wd + jq_l) * 32 + half * 2;
#pragma unroll
    for (int s = 0; s < 8; ++s) a[s] = *(const v2f*)(qrow + 4 * s);

    // ---- S = Q @ K^T : 10 key tiles ----
    v8f c[10] = {};
#pragma unroll
    for (int t = 0; t < 10; ++t) {
        const int kk = t * 16 + l16;               // key index this lane supplies
        const int kr = min(kk / 22, 6);
        const int kc = min(kk - kr * 22, 21);
        const int ki = rstart + kr;
        const int kj = min(cstart + kc, Wd - 1);
        const float* krow = Kb + (size_t)(ki * Wd + kj) * 32 + half * 2;
#pragma unroll
        for (int s = 0; s < 8; ++s) {
            v2f b = *(const v2f*)(krow + 4 * s);
            c[t] = wmma_f32(a[s], b, c[t]);
        }
    }

    // ---- mask + row softmax (rows live in one 16-lane half per vgpr) ----
    float rs[8];
#pragma unroll
    for (int r = 0; r < 8; ++r) {
        const int m  = r + half * 8;
        const int jq = min(j0 + m, Wd - 1);
        const int cq = min(max(jq - 3, 0), 49);    // this query's window start col
        float mx = -3.0e38f;
#pragma unroll
        for (int t = 0; t < 10; ++t) {
            const int kk = t * 16 + l16;
            const int kr = kk / 22;
            const int kc = kk - kr * 22;
            const int kj = cstart + kc;
            const bool valid = (kk < 154) && (kj >= cq) && (kj < cq + 7);
            const float sv = valid ? c[t][r] : -1.0e30f;
            c[t][r] = sv;
            mx = fmaxf(mx, sv);
        }
#pragma unroll
        for (int off = 1; off < 16; off <<= 1)
            mx = fmaxf(mx, __shfl_xor(mx, off, 32));
        float sum = 0.0f;
#pragma unroll
        for (int t = 0; t < 10; ++t) {
            const float p = __expf(c[t][r] - mx);
            c[t][r] = p;
            sum += p;
        }
#pragma unroll
        for (int off = 1; off < 16; off <<= 1)
            sum += __shfl_xor(sum, off, 32);
        rs[r] = sum;
    }

    // ---- stage P through LDS: C-layout -> A-layout ----
    __shared__ float pbuf[16 * 160];
#pragma unroll
    for (int t = 0; t < 10; ++t)
#pragma unroll
        for (int r = 0; r < 8; ++r)
            pbuf[(r + half * 8) * 160 + t * 16 + l16] = c[t][r];
    __syncthreads();

    // ---- O = P(16x160) @ V(160x32) ----
    v8f o[2] = {};
    for (int s = 0; s < 40; ++s) {
        const int kkb = 4 * s + half * 2;
        v2f pa = *(const v2f*)(&pbuf[l16 * 160 + kkb]);
        const int kr0 = min(kkb / 22, 6),       kc0 = min(kkb - kr0 * 22, 21);
        const int kr1 = min((kkb + 1) / 22, 6), kc1 = min(kkb + 1 - kr1 * 22, 21);
        const size_t p0 = (size_t)((rstart + kr0) * Wd + min(cstart + kc0, Wd - 1)) * 32;
        const size_t p1 = (size_t)((rstart + kr1) * Wd + min(cstart + kc1, Wd - 1)) * 32;
#pragma unroll
        for (int t = 0; t < 2; ++t) {
            const int dim = t * 16 + l16;
            v2f b;
            b.x = Vb[p0 + dim];
            b.y = Vb[p1 + dim];
            o[t] = wmma_f32(pa, b, o[t]);
        }
    }

    // ---- normalize, write (B, N, heads*32) ----
    const int b_ = bh >> 3, h_ = bh & 7;
#pragma unroll
    for (int t = 0; t < 2; ++t)
#pragma unroll
        for (int r = 0; r < 8; ++r) {
            const int m = r + half * 8;
            const int j = j0 + m;
            if (j < Wd) {
                const size_t orow =
                    ((size_t)b_ * Npix + (size_t)(i * Wd + j)) * 256 + h_ * 32 + t * 16 + l16;
                O[orow] = o[t][r] / rs[r];
            }
        }
}

// ---------------------------------------------------------------------------
extern "C" void kernel_launch(void* const* d_in, const int* in_sizes, int n_in,
                              void* d_out, int out_size, void* d_ws, size_t ws_size,
                              hipStream_t stream)
{
    const float* x     = (const float*)d_in[0];
    const float* Wqkv  = (const float*)d_in[1];
    const float* Wproj = (const float*)d_in[2];
    const float* bproj = (const float*)d_in[3];
    (void)in_sizes; (void)n_in; (void)out_size; (void)ws_size;

    const int B = 2, Hh = 56, C = 256, N = 3136, heads = 8, d = 32;
    const int BN = B * N;                       // 6272 rows

    float* ws      = (float*)d_ws;
    float* qkvraw  = ws;                                  // BN*768 floats
    float* Q       = ws + (size_t)BN * 768;               // B*h*N*d
    float* Kt      = Q  + (size_t)B * heads * N * d;
    float* V       = Kt + (size_t)B * heads * N * d;
    float* attnout = ws;                                  // alias qkvraw (dead after rope)

    // 1) QKV projection: (6272 x 256) @ (256 x 768)
    gemm16x64_f32wmma<<<dim3(768 / 64, BN / (16 * 8)), 256, 0, stream>>>(
        x, Wqkv, nullptr, qkvraw, C, 3 * C);

    // 2) RoPE + scatter to (B, h, N, d), Q pre-scaled
    const int total = BN * heads * (d / 2);
    rope_scatter<<<(total + 255) / 256, 256, 0, stream>>>(qkvraw, Q, Kt, V, BN, N);

    // 3) 7x7 neighborhood attention
    natten_wmma<<<dim3(4, Hh, B * heads), 32, 0, stream>>>(Q, Kt, V, attnout);

    // 4) Output projection + bias: (6272 x 256) @ (256 x 256)
    gemm16x64_f32wmma<<<dim3(C / 64, BN / (16 * 8)), 256, 0, stream>>>(
        attnout, Wproj, bproj, (float*)d_out, C, C);
}